// PointTransformerV3_55825984913721
// MI455X (gfx1250) — compile-verified
//
#include <hip/hip_runtime.h>
#include <hip/hip_bf16.h>

// ---------------------------------------------------------------------------
// PointTransformerV3 serialized attention for gfx1250 (MI455X).
// fp32 WMMA (V_WMMA_F32_16X16X4_F32) + async global->LDS staging (ASYNCcnt).
// ---------------------------------------------------------------------------

typedef __attribute__((ext_vector_type(2))) float v2f;
typedef __attribute__((ext_vector_type(8))) float v8f;
typedef __attribute__((ext_vector_type(4))) int   v4i;

#define WMMA_F32_4(a, b, c)                                                  \
  __builtin_amdgcn_wmma_f32_16x16x4_f32(false, (a), false, (b), (short)0,    \
                                        (c), false, false)

#define AS1 __attribute__((address_space(1)))
#define AS3 __attribute__((address_space(3)))

// Async bulk copy: 16 bytes global -> LDS, tracked by ASYNCcnt (no VGPR hop).
__device__ __forceinline__ void stage_b128(const float* g, float* l) {
#if __has_builtin(__builtin_amdgcn_global_load_async_to_lds_b128)
  __builtin_amdgcn_global_load_async_to_lds_b128((AS1 v4i*)g, (AS3 v4i*)l,
                                                 0, 0);
#else
  *(float4*)l = *(const float4*)g;
#endif
}

__device__ __forceinline__ void async_wait0() {
#if __has_builtin(__builtin_amdgcn_s_wait_asynccnt)
  __builtin_amdgcn_s_wait_asynccnt(0);
#elif __has_builtin(__builtin_amdgcn_global_load_async_to_lds_b128)
  asm volatile("s_wait_asynccnt 0" ::: "memory");
#endif
}

#define CCH   384     // channels
#define HEADS 6
#define DH    64      // head dim
#define KP    256     // patch size
#define C3    1152    // 3*C
#define RPE_ROWS 123  // 3*41
#define SLD   260     // padded LDS row stride for S  (bank-spread, 4n+k)
#define VLD   68      // padded LDS row stride for V  (16B aligned, 4n+k)
#define ALD   36      // padded LDS row stride for GEMM chunks (16B aligned)
#define QB    64      // query rows per attention workgroup

// ---------------------------------------------------------------------------
// Kernels 1/3: Y[row][col] = X[gather[row]] . W[col] + bias[col]
// 256 threads (8 waves) per 64x64 tile; K chunked by 32 through LDS with
// async b128 staging; each wave owns two 16x16 subtiles sharing a column so
// the B fragment is reused across both WMMAs.
// ---------------------------------------------------------------------------
__global__ __launch_bounds__(256)
void gemm_gather_wmma(const float* __restrict__ X,      // [src_rows, Cin]
                      const float* __restrict__ W,      // [Cout, Cin]
                      const float* __restrict__ bias,   // [Cout]
                      const long long* __restrict__ gather, // [rows]
                      float* __restrict__ Y,            // [rows, Cout]
                      int Cin, int Cout) {
  __shared__ float a_s[64 * ALD];
  __shared__ float b_s[64 * ALD];
  __shared__ long long s_gidx[64];

  const int tid  = threadIdx.x;
  const int lane = tid & 31;
  const int wave = tid >> 5;             // 0..7
  const int l15  = lane & 15;
  const int half = lane >> 4;            // A/B frag K split: 0 -> k,k+1  1 -> k+2,k+3
  const int koff = half * 2;
  const int colBase = blockIdx.x * 64;
  const int rowBase = blockIdx.y * 64;

  if (tid < 64) s_gidx[tid] = gather[rowBase + tid];

  // wave's two subtiles: t and t+8 -> same column, rows tm and tm+2
  const int tm0 = wave >> 2, tn = wave & 3;
  const int tm1 = tm0 + 2;
  v8f acc0 = {}, acc1 = {};

  for (int k0 = 0; k0 < Cin; k0 += 32) {
    __syncthreads();   // previous chunk fully consumed (also orders s_gidx)
    // stage A chunk: 64 gathered rows x 32 floats  (512 b128, 2/thread)
    for (int i = tid; i < 512; i += 256) {
      const int r = i >> 3, c = i & 7;
      stage_b128(X + s_gidx[r] * (long long)Cin + k0 + c * 4,
                 &a_s[r * ALD + c * 4]);
    }
    // stage B chunk: 64 weight rows x 32 floats
    for (int i = tid; i < 512; i += 256) {
      const int r = i >> 3, c = i & 7;
      stage_b128(W + (long long)(colBase + r) * Cin + k0 + c * 4,
                 &b_s[r * ALD + c * 4]);
    }
    async_wait0();
    __syncthreads();

#pragma unroll
    for (int kk = 0; kk < 32; kk += 4) {
      const float* ba = &b_s[(tn * 16 + l15) * ALD + kk + koff];
      v2f b = {ba[0], ba[1]};
      const float* a0 = &a_s[(tm0 * 16 + l15) * ALD + kk + koff];
      v2f av0 = {a0[0], a0[1]};
      acc0 = WMMA_F32_4(av0, b, acc0);
      const float* a1 = &a_s[(tm1 * 16 + l15) * ALD + kk + koff];
      v2f av1 = {a1[0], a1[1]};
      acc1 = WMMA_F32_4(av1, b, acc1);
    }
  }

  const float bv = bias[colBase + tn * 16 + l15];
#pragma unroll
  for (int r = 0; r < 8; ++r) {
    const int m = r + 8 * half;          // D layout: VGPR r -> M=r / M=r+8
    Y[(long long)(rowBase + tm0 * 16 + m) * Cout + colBase + tn * 16 + l15] =
        acc0[r] + bv;
    Y[(long long)(rowBase + tm1 * 16 + m) * Cout + colBase + tn * 16 + l15] =
        acc1[r] + bv;
  }
}

// ---------------------------------------------------------------------------
// Kernel 2: fused attention for one (patch, head, 64-query block).
// V is async-staged into LDS at entry; the copy overlaps the whole
// QK^T + bias + softmax phase and is only waited on before PV.
// ---------------------------------------------------------------------------
__global__ __launch_bounds__(256)
void attn_fused_wmma(const float* __restrict__ qkv,   // [N, 1152] serialized
                     const float* __restrict__ rpe,   // [123, 6]
                     const int* __restrict__ gcoord,  // [N, 3] original order
                     const long long* __restrict__ order, // [N]
                     float* __restrict__ out) {       // [N, 384] serialized
  __shared__ float s_S[QB * SLD];           // 66560 B
  __shared__ float s_v[KP * VLD];           // 69632 B
  __shared__ int   s_c[KP * 3];             //  3072 B
  __shared__ float s_rpe[RPE_ROWS * HEADS]; //  2952 B

  const int tid  = threadIdx.x;
  const int lane = tid & 31;
  const int wave = tid >> 5;              // 0..7
  const int l15  = lane & 15;
  const int half = lane >> 4;
  const int koff = half * 2;

  const int p  = blockIdx.x;              // patch
  const int h  = blockIdx.y;              // head
  const int qb = blockIdx.z;              // query block (0..3)

  const long long rowP = (long long)p * KP;
  const float* __restrict__ qbase = qkv + rowP * C3 + h * DH;
  const float* __restrict__ kbase = qkv + rowP * C3 + CCH + h * DH;
  const float* __restrict__ vbase = qkv + rowP * C3 + 2 * CCH + h * DH;

  // ---- kick off async V stage: 256 rows x 64 floats (4096 b128, 16/thr) ----
  for (int i = tid; i < KP * 16; i += 256) {
    const int r = i >> 4, c = i & 15;
    stage_b128(vbase + (long long)r * C3 + c * 4, &s_v[r * VLD + c * 4]);
  }

  // ---- stage patch coords (via serialization order) + rpe table ----
  for (int i = tid; i < KP; i += 256) {
    const long long g = order[rowP + i];
    s_c[i * 3 + 0] = gcoord[g * 3 + 0];
    s_c[i * 3 + 1] = gcoord[g * 3 + 1];
    s_c[i * 3 + 2] = gcoord[g * 3 + 2];
  }
  for (int i = tid; i < RPE_ROWS * HEADS; i += 256) s_rpe[i] = rpe[i];
  __syncthreads();

  // ---- S = Q K^T : 4x16 tiles of 16x16 over [QB x KP], 8 tiles/wave ----
  for (int t = wave; t < 64; t += 8) {
    const int tm = t >> 4;
    const int tn = t & 15;
    const float* __restrict__ aptr =
        qbase + (long long)(qb * QB + tm * 16 + l15) * C3;
    const float* __restrict__ bptr = kbase + (long long)(tn * 16 + l15) * C3;
    v8f acc = {};
#pragma unroll
    for (int k = 0; k < DH; k += 4) {
      v2f a = {aptr[k + koff], aptr[k + koff + 1]};
      v2f b = {bptr[k + koff], bptr[k + koff + 1]};
      acc = WMMA_F32_4(a, b, acc);
    }
#pragma unroll
    for (int r = 0; r < 8; ++r)
      s_S[(tm * 16 + r + 8 * half) * SLD + tn * 16 + l15] = acc[r];
  }
  __syncthreads();

  // ---- scale + relative-position bias ----
  const float scale = 0.125f;             // 64^-0.5
  for (int e = tid; e < QB * KP; e += 256) {
    const int ql = e >> 8;
    const int kk = e & 255;
    const int qg = qb * QB + ql;
    float bias = 0.f;
#pragma unroll
    for (int a = 0; a < 3; ++a) {
      int rel = s_c[qg * 3 + a] - s_c[kk * 3 + a];
      rel = min(max(rel, -20), 20) + 20 + a * 41;
      bias += s_rpe[rel * HEADS + h];
    }
    s_S[ql * SLD + kk] = s_S[ql * SLD + kk] * scale + bias;
  }
  __syncthreads();

  // ---- softmax: one 256-wide row per wave iteration ----
  for (int r = wave; r < QB; r += 8) {
    float* __restrict__ row = &s_S[r * SLD];
    float m = -3.402823466e38f;
#pragma unroll
    for (int i = 0; i < 8; ++i) m = fmaxf(m, row[lane + i * 32]);
#pragma unroll
    for (int off = 16; off > 0; off >>= 1) m = fmaxf(m, __shfl_xor(m, off, 32));
    float ex[8];
    float sum = 0.f;
#pragma unroll
    for (int i = 0; i < 8; ++i) {
      ex[i] = __expf(row[lane + i * 32] - m);
      sum += ex[i];
    }
#pragma unroll
    for (int off = 16; off > 0; off >>= 1) sum += __shfl_xor(sum, off, 32);
    const float inv = 1.f / sum;
#pragma unroll
    for (int i = 0; i < 8; ++i) row[lane + i * 32] = ex[i] * inv;
  }

  // V stage issued at entry must have landed before PV reads it.
  async_wait0();
  __syncthreads();

  // ---- O = P V : 4x4 tiles of 16x16 over [QB x DH], 2 tiles/wave ----
  for (int t = wave; t < 16; t += 8) {
    const int tm = t >> 2;
    const int tn = t & 3;
    const float* __restrict__ aptr = &s_S[(tm * 16 + l15) * SLD];
    v8f acc = {};
#pragma unroll 8
    for (int k = 0; k < KP; k += 4) {
      v2f a = {aptr[k + koff], aptr[k + koff + 1]};
      v2f b = {s_v[(k + koff) * VLD + tn * 16 + l15],
               s_v[(k + koff + 1) * VLD + tn * 16 + l15]};
      acc = WMMA_F32_4(a, b, acc);
    }
#pragma unroll
    for (int r = 0; r < 8; ++r) {
      const int qrow = qb * QB + tm * 16 + r + 8 * half;
      out[(rowP + qrow) * (long long)CCH + h * DH + tn * 16 + l15] = acc[r];
    }
  }
}

// ---------------------------------------------------------------------------
extern "C" void kernel_launch(void* const* d_in, const int* in_sizes, int n_in,
                              void* d_out, int out_size, void* d_ws, size_t ws_size,
                              hipStream_t stream) {
  const float*     feat    = (const float*)d_in[0];      // [N, 384]
  const float*     qkv_w   = (const float*)d_in[1];      // [1152, 384]
  const float*     qkv_b   = (const float*)d_in[2];      // [1152]
  const float*     proj_w  = (const float*)d_in[3];      // [384, 384]
  const float*     proj_b  = (const float*)d_in[4];      // [384]
  const float*     rpe     = (const float*)d_in[5];      // [123, 6]
  const long long* order   = (const long long*)d_in[6];  // [N] int64
  const long long* inverse = (const long long*)d_in[7];  // [N] int64
  const int*       gcoord  = (const int*)d_in[8];        // [N, 3] int32

  const int N = in_sizes[0] / CCH;        // 65536
  const int P = N / KP;                   // 256

  float* qkv_s    = (float*)d_ws;                        // [N, 1152]
  float* attn_out = qkv_s + (size_t)N * C3;              // [N, 384]

  // 1) qkv projection fused with serialization gather
  {
    dim3 grid(C3 / 64, N / 64);
    gemm_gather_wmma<<<grid, 256, 0, stream>>>(feat, qkv_w, qkv_b, order,
                                               qkv_s, CCH, C3);
  }

  // 2) fused per-patch attention (scale, RPE bias, softmax, PV)
  {
    dim3 grid(P, HEADS, KP / QB);
    attn_fused_wmma<<<grid, 256, 0, stream>>>(qkv_s, rpe, gcoord, order,
                                              attn_out);
  }

  // 3) output projection fused with inverse gather
  {
    dim3 grid(CCH / 64, N / 64);
    gemm_gather_wmma<<<grid, 256, 0, stream>>>(attn_out, proj_w, proj_b,
                                               inverse, (float*)d_out,
                                               CCH, CCH);
  }
}